// Pointnet2Backbone_45054206935141
// MI455X (gfx1250) — compile-verified
//
#include <hip/hip_runtime.h>
#include <hip/hip_bf16.h>

typedef float v2f __attribute__((ext_vector_type(2)));
typedef float v8f __attribute__((ext_vector_type(8)));

#define BATCH 16
#define NPTS  40000

// ---------------------------------------------------------------------------
// Farthest point sampling: one block per batch. dists kept in global scratch
// (L2 resident: B*N*4 = 2.5MB). Tie-break toward lower index (jnp.argmax).
// ---------------------------------------------------------------------------
__global__ void fps_kernel(const float* __restrict__ xyz, float* __restrict__ dists,
                           int N, int npoint, int* __restrict__ fidx,
                           float* __restrict__ newxyz)
{
    const int b = blockIdx.x;
    const int tid = threadIdx.x;
    const int nthr = blockDim.x;
    const float* px = xyz + (size_t)b * N * 3;
    float* dd = dists + (size_t)b * N;

    __shared__ float s_val[1024];
    __shared__ int   s_idx[1024];

    for (int n = tid; n < N; n += nthr) dd[n] = 1e10f;
    __syncthreads();

    int far = 0;
    for (int i = 0; i < npoint; ++i) {
        if (tid == 0) {
            fidx[b * npoint + i] = far;
            newxyz[((size_t)b * npoint + i) * 3 + 0] = px[far * 3 + 0];
            newxyz[((size_t)b * npoint + i) * 3 + 1] = px[far * 3 + 1];
            newxyz[((size_t)b * npoint + i) * 3 + 2] = px[far * 3 + 2];
        }
        const float cx = px[far * 3 + 0];
        const float cy = px[far * 3 + 1];
        const float cz = px[far * 3 + 2];
        float best = -1.0f; int bestn = 0x7fffffff;
        for (int n = tid; n < N; n += nthr) {
            float dx = px[n * 3 + 0] - cx;
            float dy = px[n * 3 + 1] - cy;
            float dz = px[n * 3 + 2] - cz;
            float d = dx * dx + dy * dy + dz * dz;
            float m = fminf(dd[n], d);
            dd[n] = m;
            if (m > best) { best = m; bestn = n; }
        }
        s_val[tid] = best; s_idx[tid] = bestn;
        __syncthreads();
        for (int off = nthr >> 1; off > 0; off >>= 1) {
            if (tid < off) {
                float v2 = s_val[tid + off]; int i2 = s_idx[tid + off];
                if (v2 > s_val[tid] || (v2 == s_val[tid] && i2 < s_idx[tid])) {
                    s_val[tid] = v2; s_idx[tid] = i2;
                }
            }
            __syncthreads();
        }
        far = s_idx[0];
        __syncthreads();
    }
}

// ---------------------------------------------------------------------------
// Ball query: one wave32 per (b, s). First-nsample-in-index-order selection
// via ballot + prefix popcount; pad trailing slots with first valid index.
// ---------------------------------------------------------------------------
__global__ void ball_query_kernel(const float* __restrict__ xyz,
                                  const float* __restrict__ new_xyz,
                                  int N, int npoint, float r2, int ns,
                                  int* __restrict__ gidx)
{
    const int bs = blockIdx.x;              // b*npoint + s
    const int b = bs / npoint;
    const int lane = threadIdx.x;           // 0..31 (wave32)
    const float* px = xyz + (size_t)b * N * 3;
    const float cx = new_xyz[(size_t)bs * 3 + 0];
    const float cy = new_xyz[(size_t)bs * 3 + 1];
    const float cz = new_xyz[(size_t)bs * 3 + 2];
    int* out = gidx + (size_t)bs * ns;

    int cnt = 0;
    int first = -1;
    for (int base = 0; base < N && cnt < ns; base += 32) {
        const int n = base + lane;
        bool ok = false;
        if (n < N) {
            float dx = px[n * 3 + 0] - cx;
            float dy = px[n * 3 + 1] - cy;
            float dz = px[n * 3 + 2] - cz;
            ok = (dx * dx + dy * dy + dz * dz) < r2;
        }
        const unsigned mm = (unsigned)__ballot(ok);
        if (first < 0 && mm) first = base + __ffs(mm) - 1;
        if (ok) {
            int pos = cnt + __popc(mm & ((1u << lane) - 1u));
            if (pos < ns) out[pos] = n;
        }
        cnt += __popc(mm);
    }
    if (first < 0) first = 0;
    int start = cnt < ns ? cnt : ns;
    for (int j = start + lane; j < ns; j += 32) out[j] = first;
}

// ---------------------------------------------------------------------------
// Build grouped input X[b] (C x L, row-major), C = 3 + Cf, L = npoint*ns.
// Channels 0..2: gathered xyz - center.  Channels 3..: gathered features.
// ---------------------------------------------------------------------------
__global__ void gather_kernel(const float* __restrict__ xyz,
                              const float* __restrict__ feats,
                              int N, int Cf, int FN,
                              const float* __restrict__ new_xyz,
                              const int* __restrict__ gidx,
                              int npoint, int ns,
                              float* __restrict__ X, int C, int total)
{
    int t = blockIdx.x * blockDim.x + threadIdx.x;
    if (t >= total) return;
    const int j = t % ns;
    const int s = (t / ns) % npoint;
    const int b = t / (ns * npoint);
    const int idx = gidx[t];
    const int L = npoint * ns;
    const int l = s * ns + j;
    float* xb = X + (size_t)b * C * L;
    const float* p = xyz + ((size_t)b * N + idx) * 3;
    const float* q = new_xyz + ((size_t)b * npoint + s) * 3;
    xb[(size_t)0 * L + l] = p[0] - q[0];
    xb[(size_t)1 * L + l] = p[1] - q[1];
    xb[(size_t)2 * L + l] = p[2] - q[2];
    const float* fb = feats + (size_t)b * Cf * FN;
    for (int c = 0; c < Cf; ++c)
        xb[(size_t)(3 + c) * L + l] = fb[(size_t)c * FN + idx];
}

// ---------------------------------------------------------------------------
// GEMM + ReLU via V_WMMA_F32_16X16X4_F32. One wave computes a 16x64 output
// strip (4 tiles along L):
//   - A fragment reused by 4 WMMAs -> 4x arithmetic intensity on W
//   - 4 independent accumulator chains hide load latency
//   - running pointers (wr / xc) instead of per-iteration k*L index math,
//     so the inner loop has only 64-bit adds, no 64-bit multiplies
//   - ragged K (C = 6/323/643) handled by a single guarded tail, so the main
//     loop has NO divergent exec-mask branches around its loads.
// Layouts per ISA 7.12.2:
//   A (16x4): lanes 0-15 -> M=lane, VGPR0=K0/VGPR1=K1; lanes 16-31 -> K2/K3.
//   B (4x16): mirrored (N on lanes, K on VGPR/half).
//   C/D:      VGPR r: lanes 0-15 -> M=r, lanes 16-31 -> M=r+8; N=lane%16.
// ---------------------------------------------------------------------------
__global__ void gemm_relu_wmma(const float* __restrict__ W,
                               const float* __restrict__ X,
                               float* __restrict__ Y,
                               int O, int C, int L)
{
    const int l0 = blockIdx.x * 64;
    const int o0 = blockIdx.y * 16;
    const int b  = blockIdx.z;
    const float* Xb = X + (size_t)b * C * L;
    float* Yb = Y + (size_t)b * O * L;

    const int lane = threadIdx.x & 31;
    const int half = lane >> 4;
    const int tm   = lane & 15;

    v8f acc0 = {0.f,0.f,0.f,0.f,0.f,0.f,0.f,0.f};
    v8f acc1 = acc0, acc2 = acc0, acc3 = acc0;

    const float* wrow = W + (size_t)(o0 + tm) * C;
    const int C4 = C & ~3;

    // running pointers for K-row (ka = k + 2*half)
    const float* wr = wrow + 2 * half;
    const float* xc = Xb + (size_t)(2 * half) * L + l0 + tm;
    const size_t xstep = (size_t)4 * L;

    int k = 0;
    for (; k < C4; k += 4) {
        v2f a;
        a.x = wr[0];
        a.y = wr[1];
        v2f b0, b1, b2, b3;
        b0.x = xc[0];  b0.y = xc[L];
        b1.x = xc[16]; b1.y = xc[L + 16];
        b2.x = xc[32]; b2.y = xc[L + 32];
        b3.x = xc[48]; b3.y = xc[L + 48];
        acc0 = __builtin_amdgcn_wmma_f32_16x16x4_f32(false, a, false, b0, (short)0, acc0, false, false);
        acc1 = __builtin_amdgcn_wmma_f32_16x16x4_f32(false, a, false, b1, (short)0, acc1, false, false);
        acc2 = __builtin_amdgcn_wmma_f32_16x16x4_f32(false, a, false, b2, (short)0, acc2, false, false);
        acc3 = __builtin_amdgcn_wmma_f32_16x16x4_f32(false, a, false, b3, (short)0, acc3, false, false);
        wr += 4;
        xc += xstep;
    }
    if (k < C) {        // single ragged tail (C % 4 = 1..3)
        const int ka = k + 2 * half;
        const int kb = ka + 1;
        const bool oka = ka < C, okb = kb < C;
        v2f a;
        a.x = oka ? wr[0] : 0.f;
        a.y = okb ? wr[1] : 0.f;
        v2f b0, b1, b2, b3;
        b0.x = oka ? xc[0]  : 0.f;  b0.y = okb ? xc[L]      : 0.f;
        b1.x = oka ? xc[16] : 0.f;  b1.y = okb ? xc[L + 16] : 0.f;
        b2.x = oka ? xc[32] : 0.f;  b2.y = okb ? xc[L + 32] : 0.f;
        b3.x = oka ? xc[48] : 0.f;  b3.y = okb ? xc[L + 48] : 0.f;
        acc0 = __builtin_amdgcn_wmma_f32_16x16x4_f32(false, a, false, b0, (short)0, acc0, false, false);
        acc1 = __builtin_amdgcn_wmma_f32_16x16x4_f32(false, a, false, b1, (short)0, acc1, false, false);
        acc2 = __builtin_amdgcn_wmma_f32_16x16x4_f32(false, a, false, b2, (short)0, acc2, false, false);
        acc3 = __builtin_amdgcn_wmma_f32_16x16x4_f32(false, a, false, b3, (short)0, acc3, false, false);
    }

#pragma unroll
    for (int r = 0; r < 8; ++r) {
        const int row = o0 + r + 8 * half;
        float* yr = Yb + (size_t)row * L + l0 + tm;
        float v0 = acc0[r], v1 = acc1[r], v2 = acc2[r], v3 = acc3[r];
        yr[0]  = v0 > 0.f ? v0 : 0.f;
        yr[16] = v1 > 0.f ? v1 : 0.f;
        yr[32] = v2 > 0.f ? v2 : 0.f;
        yr[48] = v3 > 0.f ? v3 : 0.f;
    }
}

// ---------------------------------------------------------------------------
// Max-pool over the ns axis:  out[b, coff+o, s] = max_j Y[b, o, s*ns + j]
// (Also used for the global SA3 pool with npoint = 1.)
// ---------------------------------------------------------------------------
__global__ void maxpool_kernel(const float* __restrict__ Y, int O, int npoint,
                               int ns, float* __restrict__ out, int Ctot,
                               int coff, int total)
{
    int t = blockIdx.x * blockDim.x + threadIdx.x;
    if (t >= total) return;
    const int s = t % npoint;
    const int o = (t / npoint) % O;
    const int b = t / (npoint * O);
    const float* yb = Y + ((size_t)b * O + o) * ((size_t)npoint * ns) + (size_t)s * ns;
    float m = yb[0];
    for (int j = 1; j < ns; ++j) m = fmaxf(m, yb[j]);
    out[((size_t)b * Ctot + coff + o) * npoint + s] = m;
}

// ---------------------------------------------------------------------------
// SA3 input build: X[b, c, p] = (c < 3) ? xyz2[b, p, c] : f2[b, c-3, p]
// ---------------------------------------------------------------------------
__global__ void sa3_build_kernel(const float* __restrict__ xyz2,
                                 const float* __restrict__ f2,
                                 float* __restrict__ X, int np, int Cf,
                                 int C, int total)
{
    int t = blockIdx.x * blockDim.x + threadIdx.x;
    if (t >= total) return;
    const int p = t % np;
    const int c = (t / np) % C;
    const int b = t / (np * C);
    float v = (c < 3) ? xyz2[((size_t)b * np + p) * 3 + c]
                      : f2[((size_t)b * Cf + (c - 3)) * np + p];
    X[((size_t)b * C + c) * np + p] = v;
}

// ---------------------------------------------------------------------------
static inline int cdiv(int a, int b) { return (a + b - 1) / b; }

extern "C" void kernel_launch(void* const* d_in, const int* in_sizes, int n_in,
                              void* d_out, int out_size, void* d_ws, size_t ws_size,
                              hipStream_t stream)
{
    const float* xyz_gt   = (const float*)d_in[0];   // (B, 40000, 3)
    const float* feats_gt = (const float*)d_in[1];   // (B, 3, 40000)
    // sa1_w: d_in[2..10], sa2_w: d_in[11..19], sa3_w: d_in[20..22]

    float* out = (float*)d_out;
    float* out_xyz1 = out;                               // (B,128,3)   6144
    float* out_f1   = out + 6144;                        // (B,320,128) 655360
    float* out_xyz2 = out + 6144 + 655360;               // (B,64,3)    3072
    float* out_f2   = out + 6144 + 655360 + 3072;        // (B,640,64)  655360
    float* out_f3   = out + 6144 + 655360 + 3072 + 655360; // (B,128)   2048

    // ---- workspace layout ----
    char* ws = (char*)d_ws;
    size_t off = 0;
    float* d_dists = (float*)(ws + off); off += (size_t)BATCH * NPTS * sizeof(float);
    int*   d_fidx  = (int*)(ws + off);   off += (size_t)BATCH * 128 * sizeof(int);
    int*   d_gidx  = (int*)(ws + off);   off += (size_t)BATCH * 128 * 64 * sizeof(int);
    const size_t BUF_ELEMS = (size_t)BATCH * 128 * 8192;   // largest O*L per batch
    float* bufA = (float*)(ws + off); off += BUF_ELEMS * sizeof(float);
    float* bufB = (float*)(ws + off); off += BUF_ELEMS * sizeof(float);
    (void)ws_size; (void)in_sizes; (void)n_in; (void)out_size;

    static const int   SA1_NS[3] = {16, 32, 64};
    static const float SA1_R[3]  = {0.1f, 0.2f, 0.4f};
    static const int   SA1_D[3][4] = {{6,32,32,64},{6,64,64,128},{6,64,96,128}};
    static const int   SA1_COFF[3] = {0, 64, 192};         // into 320 channels

    static const int   SA2_NS[3] = {8, 16, 32};
    static const float SA2_R[3]  = {0.2f, 0.4f, 0.8f};
    static const int   SA2_D[3][4] = {{323,64,64,128},{323,128,128,256},{323,128,128,256}};
    static const int   SA2_COFF[3] = {0, 128, 384};        // into 640 channels

    static const int   SA3_D[4] = {643, 256, 512, 128};

    // ======================= SA1 =======================
    fps_kernel<<<BATCH, 1024, 0, stream>>>(xyz_gt, d_dists, NPTS, 128, d_fidx, out_xyz1);

    for (int r = 0; r < 3; ++r) {
        const int ns = SA1_NS[r];
        const int L  = 128 * ns;
        const float r2 = SA1_R[r] * SA1_R[r];
        ball_query_kernel<<<BATCH * 128, 32, 0, stream>>>(
            xyz_gt, out_xyz1, NPTS, 128, r2, ns, d_gidx);
        {
            int total = BATCH * 128 * ns;
            gather_kernel<<<cdiv(total, 256), 256, 0, stream>>>(
                xyz_gt, feats_gt, NPTS, 3, NPTS, out_xyz1, d_gidx,
                128, ns, bufA, 6, total);
        }
        float* src = bufA; float* dst = bufB;
        for (int l = 0; l < 3; ++l) {
            const int Cd = SA1_D[r][l], Od = SA1_D[r][l + 1];
            const float* Wp = (const float*)d_in[2 + 3 * r + l];
            dim3 grid(L / 64, Od / 16, BATCH);
            gemm_relu_wmma<<<grid, 32, 0, stream>>>(Wp, src, dst, Od, Cd, L);
            float* t = src; src = dst; dst = t;
        }
        // final activations are in `src` after the swap
        const int Olast = SA1_D[r][3];
        int total = BATCH * Olast * 128;
        maxpool_kernel<<<cdiv(total, 256), 256, 0, stream>>>(
            src, Olast, 128, ns, out_f1, 320, SA1_COFF[r], total);
    }

    // ======================= SA2 =======================
    fps_kernel<<<BATCH, 1024, 0, stream>>>(out_xyz1, d_dists, 128, 64, d_fidx, out_xyz2);

    for (int r = 0; r < 3; ++r) {
        const int ns = SA2_NS[r];
        const int L  = 64 * ns;
        const float r2 = SA2_R[r] * SA2_R[r];
        ball_query_kernel<<<BATCH * 64, 32, 0, stream>>>(
            out_xyz1, out_xyz2, 128, 64, r2, ns, d_gidx);
        {
            int total = BATCH * 64 * ns;
            gather_kernel<<<cdiv(total, 256), 256, 0, stream>>>(
                out_xyz1, out_f1, 128, 320, 128, out_xyz2, d_gidx,
                64, ns, bufA, 323, total);
        }
        float* src = bufA; float* dst = bufB;
        for (int l = 0; l < 3; ++l) {
            const int Cd = SA2_D[r][l], Od = SA2_D[r][l + 1];
            const float* Wp = (const float*)d_in[11 + 3 * r + l];
            dim3 grid(L / 64, Od / 16, BATCH);
            gemm_relu_wmma<<<grid, 32, 0, stream>>>(Wp, src, dst, Od, Cd, L);
            float* t = src; src = dst; dst = t;
        }
        const int Olast = SA2_D[r][3];
        int total = BATCH * Olast * 64;
        maxpool_kernel<<<cdiv(total, 256), 256, 0, stream>>>(
            src, Olast, 64, ns, out_f2, 640, SA2_COFF[r], total);
    }

    // ======================= SA3 (global) =======================
    {
        const int np = 64, C0 = 643;
        int total = BATCH * C0 * np;
        sa3_build_kernel<<<cdiv(total, 256), 256, 0, stream>>>(
            out_xyz2, out_f2, bufA, np, 640, C0, total);

        float* src = bufA; float* dst = bufB;
        for (int l = 0; l < 3; ++l) {
            const int Cd = SA3_D[l], Od = SA3_D[l + 1];
            const float* Wp = (const float*)d_in[20 + l];
            dim3 grid(np / 64, Od / 16, BATCH);
            gemm_relu_wmma<<<grid, 32, 0, stream>>>(Wp, src, dst, Od, Cd, np);
            float* t = src; src = dst; dst = t;
        }
        int mtotal = BATCH * 128 * 1;
        maxpool_kernel<<<cdiv(mtotal, 256), 256, 0, stream>>>(
            src, 128, 1, 64, out_f3, 128, 0, mtotal);
    }
}